// Downsample_PASA_29368986370267
// MI455X (gfx1250) — compile-verified
//
#include <hip/hip_runtime.h>

typedef __attribute__((ext_vector_type(2))) float v2f;
typedef __attribute__((ext_vector_type(4))) float v4f;
typedef __attribute__((ext_vector_type(8))) float v8f;

// Only check the builtin on the device pass; the host pass doesn't register
// AMDGPU target builtins for __has_builtin.
#if defined(__HIP_DEVICE_COMPILE__) && !__has_builtin(__builtin_amdgcn_wmma_f32_16x16x4_f32)
#error "builtin __builtin_amdgcn_wmma_f32_16x16x4_f32 not available on device"
#endif

#define HH 112
#define WW 112
#define CIN 128
#define OHH 56
#define OWW 56
#define TCOLS 113          // input cols incl. left reflect pad (-1..111)
#define CHUNK 32           // channels staged in LDS at a time
#define CPAD 36            // channel-minor stride (mult of 4 for b128 loads)
#define TILE_F (3*TCOLS*CPAD)   // 12204 floats
#define SIG_F (4*16*9)          // sigma buffer: 4 waves x 16 pos x 9
#define A_FLOATS (288*32*2)     // packed WMMA A operand in d_ws

// ---------------------------------------------------------------------------
// Prep: fold BN scale into conv weights and pack into the WMMA-A lane layout.
// K order: k = t*128 + c  (t = 3*di+dj tap index, c = input channel).
// Step s (0..287): t = s/32, c_abs = 4*(s%32).
// Lane l: M = l%16 (0 for M>=9), channels c_abs + 2*(l/16) + {0,1} -> float2.
// Also writes shift[9] at ws[A_FLOATS].
// ---------------------------------------------------------------------------
__global__ void pasa_prep(const float* __restrict__ w,
                          const float* __restrict__ bw,
                          const float* __restrict__ bb,
                          const float* __restrict__ bm,
                          const float* __restrict__ bv,
                          float* __restrict__ ws) {
  int e = blockIdx.x * 256 + threadIdx.x;   // 0..9215
  if (e < 288 * 32) {
    int s = e >> 5, l = e & 31;
    int t = s >> 5;
    int c = ((s & 31) << 2) + ((l >> 4) << 1);
    int m = l & 15;
    int di = t / 3, dj = t - 3 * di;
    v2f a; a.x = 0.f; a.y = 0.f;
    if (m < 9) {
      float scale = bw[m] * rsqrtf(bv[m] + 1e-5f);
      a.x = w[((m * CIN + c) * 3 + di) * 3 + dj] * scale;
      a.y = w[((m * CIN + c + 1) * 3 + di) * 3 + dj] * scale;
    }
    ((v2f*)ws)[e] = a;
  }
  if (blockIdx.x == 0 && threadIdx.x < 9) {
    int m = threadIdx.x;
    float scale = bw[m] * rsqrtf(bv[m] + 1e-5f);
    ws[A_FLOATS + m] = bb[m] - bm[m] * scale;
  }
}

// ---------------------------------------------------------------------------
// Main: one block per (batch n, output row oh). 128 threads = 4 waves.
// Wave w owns 16 output positions ow = 16w + lane%16 (clamped; dup stores of
// identical values are benign). Sigma via WMMA f32 16x16x4 over K=1152,
// accumulated across 4 LDS-staged channel chunks; then normalize and do the
// 9-tap sigma-weighted aggregation per channel from LDS.
// ---------------------------------------------------------------------------
__global__ __launch_bounds__(128) void pasa_main(const float* __restrict__ x,
                                                 const float* __restrict__ ws,
                                                 float* __restrict__ out) {
  __shared__ float smem[TILE_F + SIG_F];
  float* sigb = smem + TILE_F;

  const int tid  = threadIdx.x;
  const int lane = tid & 31;
  const int wv   = tid >> 5;
  const int p    = lane & 15;
  const int half = lane >> 4;
  const int oh   = blockIdx.x;   // 0..55
  const int n    = blockIdx.y;   // 0..7
  int ow = wv * 16 + p; if (ow > OWW - 1) ow = OWW - 1;
  const int r0 = 2 * oh - 1;     // top input row of the 3x3 window band

  const float* xn    = x + (size_t)n * CIN * HH * WW;
  const v2f*  Apack  = (const v2f*)ws;
  const float* shift = ws + A_FLOATS;

  // Stage 3 rows x 113 cols x CHUNK channels, channel-minor [row][col][ch].
  // Consecutive threads -> consecutive cols: coalesced global, stride-CPAD LDS.
  auto stage = [&](int cb) {
    for (int e = tid; e < CHUNK * 3 * TCOLS; e += 128) {
      int col = e % TCOLS;
      int rc  = e / TCOLS;
      int r   = rc % 3;
      int cl  = rc / 3;
      int gr = r0 + r;  if (gr < 0) gr = 1;   // reflect (index -1 -> 1)
      int gc = col - 1; if (gc < 0) gc = 1;   // reflect
      smem[(r * TCOLS + col) * CPAD + cl] = xn[((cb + cl) * HH + gr) * WW + gc];
    }
  };

  int bofs[9];
  #pragma unroll
  for (int t = 0; t < 9; ++t) {
    int di = t / 3, dj = t - 3 * di;
    bofs[t] = (di * TCOLS + 2 * ow + dj) * CPAD;
  }

  // ---- sigma accumulation over 4 channel chunks ----
  v8f acc = {0.f, 0.f, 0.f, 0.f, 0.f, 0.f, 0.f, 0.f};
  for (int pass = 0; pass < 4; ++pass) {
    const int cb = pass * CHUNK;
    if (pass) __syncthreads();
    stage(cb);
    __syncthreads();
    #pragma unroll
    for (int t = 0; t < 9; ++t) {
      const int sbase = t * 32 + (cb >> 2);
      const int bb_   = bofs[t] + (half << 1);
      #pragma unroll
      for (int cs = 0; cs < 8; ++cs) {
        v2f a = Apack[(sbase + cs) * 32 + lane];            // global b64, coalesced
        v2f b = *(const v2f*)(smem + bb_ + (cs << 2));      // ds b64
        acc = __builtin_amdgcn_wmma_f32_16x16x4_f32(
            false, a, false, b, (short)0, acc, false, false);
      }
    }
  }

  // ---- sigma finalize: acc lane layout -> LDS, shift+clamp+normalize ----
  #pragma unroll
  for (int j = 0; j < 8; ++j) {
    int m = j + (half << 3);                 // lanes 0-15: M=j, lanes 16-31: M=j+8
    if (m < 9) sigb[(wv * 16 + p) * 9 + m] = acc[j];
  }
  if (half == 0) {                           // intra-wave LDS ops are in-order
    float sg[9]; float sum = 0.f;
    #pragma unroll
    for (int m = 0; m < 9; ++m) {
      float v = sigb[(wv * 16 + p) * 9 + m] + shift[m];
      v = v < 1e-4f ? 1e-4f : v;
      sg[m] = v; sum += v;
    }
    float inv = 1.0f / sum;
    #pragma unroll
    for (int m = 0; m < 9; ++m) sigb[(wv * 16 + p) * 9 + m] = sg[m] * inv;
  }
  __syncthreads();

  float s9[9];
  #pragma unroll
  for (int t = 0; t < 9; ++t) s9[t] = sigb[(wv * 16 + p) * 9 + t];

  // ---- aggregation: out[c] = sum_t s9[t] * x(tap t, c), 16 ch per half-wave
  auto aggregate = [&](int cb) {
    const int clb = half * 16;
    #pragma unroll
    for (int cc = 0; cc < 16; cc += 4) {
      v4f sum = {0.f, 0.f, 0.f, 0.f};
      #pragma unroll
      for (int t = 0; t < 9; ++t) {
        v4f xv = *(const v4f*)(smem + bofs[t] + clb + cc);  // ds b128
        sum += xv * s9[t];
      }
      float* op = out + (((size_t)n * CIN + (cb + clb + cc)) * OHH + oh) * OWW + ow;
      op[0]             = sum[0];
      op[OHH * OWW]     = sum[1];
      op[2 * OHH * OWW] = sum[2];
      op[3 * OHH * OWW] = sum[3];
    }
  };

  aggregate(96);                              // chunk 96..127 is still in LDS
  for (int pass = 0; pass < 3; ++pass) {      // restage 0,32,64 (L2-resident)
    __syncthreads();
    stage(pass * CHUNK);
    __syncthreads();
    aggregate(pass * CHUNK);
  }
}

// ---------------------------------------------------------------------------
extern "C" void kernel_launch(void* const* d_in, const int* in_sizes, int n_in,
                              void* d_out, int out_size, void* d_ws, size_t ws_size,
                              hipStream_t stream) {
  const float* x  = (const float*)d_in[0];
  const float* w  = (const float*)d_in[1];
  const float* bw = (const float*)d_in[2];
  const float* bb = (const float*)d_in[3];
  const float* bm = (const float*)d_in[4];
  const float* bv = (const float*)d_in[5];
  float* ws  = (float*)d_ws;
  float* out = (float*)d_out;

  hipLaunchKernelGGL(pasa_prep, dim3(36), dim3(256), 0, stream,
                     w, bw, bb, bm, bv, ws);
  hipLaunchKernelGGL(pasa_main, dim3(OHH, 8), dim3(128), 0, stream,
                     x, ws, out);
}